// Butterfly_11982958756783
// MI455X (gfx1250) — compile-verified
//
#include <hip/hip_runtime.h>

typedef float v2f __attribute__((ext_vector_type(2)));
typedef float v4f __attribute__((ext_vector_type(4)));
typedef float v8f __attribute__((ext_vector_type(8)));

// LDS tile: ROWS batch rows x 1024 elements, stride padded to 1028 floats
// (multiple of 4 for b128 alignment; 1028 % 64 == 4 -> conflict-friendly).
constexpr int T_STRIDE = 1028;
constexpr int ROWS = 32;                        // batch rows per workgroup
constexpr int LDS_BYTES = ROWS * T_STRIDE * 4;  // 131584 B -> 2 WGs per WGP

// ---------------------------------------------------------------------------
// Kernel 1: fold the 10 butterfly stages into 64 dense 32x32 matrices.
//   mats[0   .. 32*1024)  : M1[b] = S4*S3*S2*S1*S0 restricted to block b
//   mats[32K .. 64*1024)  : M2[u] = S9*S8*S7*S6*S5 restricted to residue u
// Stage s pairs elements differing by 2^s; twiddle pair index for element n is
// p = ((n >> (s+1)) << s) | (n & (2^s - 1)); output row i = bit s of n.
// ---------------------------------------------------------------------------
__global__ void butterfly_build_mats(const float* __restrict__ tw,
                                     float* __restrict__ mats) {
  __shared__ float Msh[32][33];
  const int c = threadIdx.x;   // column of the dense matrix
  const int a = threadIdx.y;   // row (element within group)
  const int m = blockIdx.x;    // 0..63
  const int phase = m >> 5;
  const int g = m & 31;        // block b (phase 0) or residue u (phase 1)

  Msh[a][c] = (a == c) ? 1.0f : 0.0f;
  __syncthreads();

  if (phase == 0) {
    // n = 32*g + a, stages s = 0..4  ->  p = 16*g + p_a, all local to block g
    for (int s = 0; s < 5; ++s) {
      const int bit = 1 << s;
      const int i = (a >> s) & 1;
      const int p = 16 * g + (((a >> (s + 1)) << s) | (a & (bit - 1)));
      const float t0 = tw[s * 2048 + p * 4 + i * 2 + 0];
      const float t1 = tw[s * 2048 + p * 4 + i * 2 + 1];
      const float v = t0 * Msh[a & ~bit][c] + t1 * Msh[a | bit][c];
      __syncthreads();
      Msh[a][c] = v;
      __syncthreads();
    }
  } else {
    // n = 32*a + g, stages s = 5..9  ->  p = 32*p_t + g, local to residue g
    for (int s = 5; s < 10; ++s) {
      const int sb = s - 5;
      const int bit = 1 << sb;
      const int i = (a >> sb) & 1;
      const int pt = ((a >> (sb + 1)) << sb) | (a & (bit - 1));
      const int p = 32 * pt + g;
      const float t0 = tw[s * 2048 + p * 4 + i * 2 + 0];
      const float t1 = tw[s * 2048 + p * 4 + i * 2 + 1];
      const float v = t0 * Msh[a & ~bit][c] + t1 * Msh[a | bit][c];
      __syncthreads();
      Msh[a][c] = v;
      __syncthreads();
    }
  }
  mats[m * 1024 + a * 32 + c] = Msh[a][c];
}

// ---------------------------------------------------------------------------
// Kernel 2: apply Phase1 (contiguous 32-blocks) then Phase2 (stride-32 groups)
// to a 32-row batch tile, entirely in LDS, using V_WMMA_F32_16X16X4_F32.
// Each A fragment (dense-matrix tile) is reused across TWO batch sub-tiles to
// halve L2 matrix traffic. D(16x16) = sum of 8 k-steps of A(16x4) x B(4x16).
// ---------------------------------------------------------------------------
__global__ void __launch_bounds__(256)
butterfly_wmma(const float* __restrict__ x, const float* __restrict__ bias,
               const float* __restrict__ mats, float* __restrict__ out) {
  extern __shared__ float T[];  // [ROWS][T_STRIDE]
  const int tid = threadIdx.x;
  const int row0 = blockIdx.x * ROWS;

  // --- coalesced load of the 32x1024 x-tile into LDS ------------------------
  for (int i = 0; i < ROWS; ++i) {
    const int chunk = tid + 256 * i;
    const int r = chunk >> 8;
    const int c4 = (chunk & 255) << 2;
    v4f v = *(const v4f*)(x + (size_t)(row0 + r) * 1024 + c4);
    *(v4f*)(&T[r * T_STRIDE + c4]) = v;
  }
  __syncthreads();

  const int lane = tid & 31;
  const int wv = tid >> 5;      // wave 0..7
  const int half = lane >> 4;   // lane half selects K/M sub-slice (ISA layout)
  const int l16 = lane & 15;

  // --- Phase 1: Y[32b..32b+32) = M1[b] @ X[32b..32b+32)  (in place) ---------
  const float* M1 = mats;
  for (int t = 0; t < 4; ++t) {
    const int b = wv * 4 + t;
    v2f B0[8], B1[8];
#pragma unroll
    for (int kk = 0; kk < 8; ++kk) {
      const int col = 32 * b + 4 * kk + 2 * half;   // K = 2*half + v
      B0[kk] = *(const v2f*)(&T[(0 * 16 + l16) * T_STRIDE + col]);
      B1[kk] = *(const v2f*)(&T[(1 * 16 + l16) * T_STRIDE + col]);
    }
    v8f a00 = {0, 0, 0, 0, 0, 0, 0, 0}, a01 = {0, 0, 0, 0, 0, 0, 0, 0};
    v8f a10 = {0, 0, 0, 0, 0, 0, 0, 0}, a11 = {0, 0, 0, 0, 0, 0, 0, 0};
    const float* Mb = M1 + b * 1024;
#pragma unroll
    for (int kk = 0; kk < 8; ++kk) {
      const int colA = 4 * kk + 2 * half;
      v2f A0 = *(const v2f*)(Mb + (0 * 16 + l16) * 32 + colA);
      v2f A1 = *(const v2f*)(Mb + (1 * 16 + l16) * 32 + colA);
      a00 = __builtin_amdgcn_wmma_f32_16x16x4_f32(false, A0, false, B0[kk],
                                                  (short)0, a00, false, false);
      a01 = __builtin_amdgcn_wmma_f32_16x16x4_f32(false, A1, false, B0[kk],
                                                  (short)0, a01, false, false);
      a10 = __builtin_amdgcn_wmma_f32_16x16x4_f32(false, A0, false, B1[kk],
                                                  (short)0, a10, false, false);
      a11 = __builtin_amdgcn_wmma_f32_16x16x4_f32(false, A1, false, B1[kk],
                                                  (short)0, a11, false, false);
    }
#pragma unroll
    for (int r = 0; r < 8; ++r) {
      T[(0 * 16 + l16) * T_STRIDE + 32 * b + 0 + r + 8 * half] = a00[r];
      T[(0 * 16 + l16) * T_STRIDE + 32 * b + 16 + r + 8 * half] = a01[r];
      T[(1 * 16 + l16) * T_STRIDE + 32 * b + 0 + r + 8 * half] = a10[r];
      T[(1 * 16 + l16) * T_STRIDE + 32 * b + 16 + r + 8 * half] = a11[r];
    }
  }
  __syncthreads();

  // --- Phase 2: Z[u::32] = M2[u] @ Y[u::32]  (in place) ---------------------
  const float* M2 = mats + 32 * 1024;
  for (int t = 0; t < 4; ++t) {
    const int u = wv * 4 + t;
    v2f B0[8], B1[8];
#pragma unroll
    for (int kk = 0; kk < 8; ++kk) {
      const int k0 = 4 * kk + 2 * half;
      v2f b0 = {T[(0 * 16 + l16) * T_STRIDE + u + 32 * k0],
                T[(0 * 16 + l16) * T_STRIDE + u + 32 * (k0 + 1)]};
      v2f b1 = {T[(1 * 16 + l16) * T_STRIDE + u + 32 * k0],
                T[(1 * 16 + l16) * T_STRIDE + u + 32 * (k0 + 1)]};
      B0[kk] = b0;
      B1[kk] = b1;
    }
    v8f a00 = {0, 0, 0, 0, 0, 0, 0, 0}, a01 = {0, 0, 0, 0, 0, 0, 0, 0};
    v8f a10 = {0, 0, 0, 0, 0, 0, 0, 0}, a11 = {0, 0, 0, 0, 0, 0, 0, 0};
    const float* Mu = M2 + u * 1024;
#pragma unroll
    for (int kk = 0; kk < 8; ++kk) {
      const int colA = 4 * kk + 2 * half;
      v2f A0 = *(const v2f*)(Mu + (0 * 16 + l16) * 32 + colA);
      v2f A1 = *(const v2f*)(Mu + (1 * 16 + l16) * 32 + colA);
      a00 = __builtin_amdgcn_wmma_f32_16x16x4_f32(false, A0, false, B0[kk],
                                                  (short)0, a00, false, false);
      a01 = __builtin_amdgcn_wmma_f32_16x16x4_f32(false, A1, false, B0[kk],
                                                  (short)0, a01, false, false);
      a10 = __builtin_amdgcn_wmma_f32_16x16x4_f32(false, A0, false, B1[kk],
                                                  (short)0, a10, false, false);
      a11 = __builtin_amdgcn_wmma_f32_16x16x4_f32(false, A1, false, B1[kk],
                                                  (short)0, a11, false, false);
    }
#pragma unroll
    for (int r = 0; r < 8; ++r) {
      T[(0 * 16 + l16) * T_STRIDE + u + 32 * (0 + r + 8 * half)] = a00[r];
      T[(0 * 16 + l16) * T_STRIDE + u + 32 * (16 + r + 8 * half)] = a01[r];
      T[(1 * 16 + l16) * T_STRIDE + u + 32 * (0 + r + 8 * half)] = a10[r];
      T[(1 * 16 + l16) * T_STRIDE + u + 32 * (16 + r + 8 * half)] = a11[r];
    }
  }
  __syncthreads();

  // --- coalesced store with fused bias add ----------------------------------
  for (int i = 0; i < ROWS; ++i) {
    const int chunk = tid + 256 * i;
    const int r = chunk >> 8;
    const int c4 = (chunk & 255) << 2;
    v4f v = *(const v4f*)(&T[r * T_STRIDE + c4]);
    v4f bv = *(const v4f*)(bias + c4);
    v += bv;
    *(v4f*)(out + (size_t)(row0 + r) * 1024 + c4) = v;
  }
}

// ---------------------------------------------------------------------------
extern "C" void kernel_launch(void* const* d_in, const int* in_sizes, int n_in,
                              void* d_out, int out_size, void* d_ws,
                              size_t ws_size, hipStream_t stream) {
  (void)in_sizes; (void)n_in; (void)out_size; (void)ws_size;
  const float* x = (const float*)d_in[0];      // 16384 x 1024
  const float* tw = (const float*)d_in[1];     // (1,10,512,2,2)
  const float* bias = (const float*)d_in[2];   // 1024
  float* out = (float*)d_out;
  float* mats = (float*)d_ws;                  // 64 * 32 * 32 floats = 256 KB

  butterfly_build_mats<<<dim3(64), dim3(32, 32), 0, stream>>>(tw, mats);
  butterfly_wmma<<<dim3(16384 / ROWS), dim3(256), LDS_BYTES, stream>>>(
      x, bias, mats, out);
}